// VectorizedHypertoroidalSplat_31817117729230
// MI455X (gfx1250) — compile-verified
//
#include <hip/hip_runtime.h>
#include <hip/hip_bf16.h>
#include <math.h>

typedef __attribute__((ext_vector_type(2))) float v2f;
typedef __attribute__((ext_vector_type(4))) float v4f;
typedef __attribute__((ext_vector_type(8))) float v8f;

#define BB   4
#define NQ   4096
#define NK   4096
#define DIMS 64
#define NFEAT 8          // 5 used, padded to 8 for two K=4 WMMAs
#define KT_GROUP 16      // k-tiles handled per wave (A-fragment reuse)

// ---------------------------------------------------------------------------
// Phase 1: per-point feature vectors.
//   Fq[b*NQ+q][0..7] = [amp*Sq, amp*w0*E0 .. amp*w3*E3, 0,0,0]
//   Fk[b*NK+k][0..7] = [Sk,     E0 .. E3,              0,0,0]
// ---------------------------------------------------------------------------
__global__ __launch_bounds__(256) void hts_features(
    const float* __restrict__ q, const float* __restrict__ k,
    const float* __restrict__ center,
    const float* __restrict__ outer_radius, const float* __restrict__ hole_radius,
    const float* __restrict__ entry, const float* __restrict__ strengths,
    const float* __restrict__ amplitude,
    float* __restrict__ Fq, float* __restrict__ Fk)
{
    const int nq_total = BB * NQ;
    const int total    = nq_total + BB * NK;
    int gid = blockIdx.x * blockDim.x + threadIdx.x;
    if (gid >= total) return;

    const bool isQ = gid < nq_total;
    const int  idx = isQ ? gid : gid - nq_total;
    const float* __restrict__ p = (isQ ? q : k) + (size_t)idx * DIMS;

    float acc_c = 0.f, a0 = 0.f, a1 = 0.f, a2 = 0.f, a3 = 0.f;
#pragma unroll 8
    for (int d = 0; d < DIMS; ++d) {
        float x  = p[d];
        float dc = x - center[d];          acc_c += dc * dc;
        float e0 = x - entry[0*DIMS + d];  a0    += e0 * e0;
        float e1 = x - entry[1*DIMS + d];  a1    += e1 * e1;
        float e2 = x - entry[2*DIMS + d];  a2    += e2 * e2;
        float e3 = x - entry[3*DIMS + d];  a3    += e3 * e3;
    }

    // torus surface distance
    float outer = fabsf(outer_radius[0]) + 1e-8f;
    float hole  = fabsf(hole_radius[0]);
    float ratio = hole / outer;
    float dist  = sqrtf(acc_c);
    float major = fabsf(dist - outer);
    float torus = fabsf(sqrtf(major * major + hole * hole) - hole);
    float surf  = (ratio < 0.1f) ? major : torus;
    float S     = expf(-0.5f * surf * surf);

    // entry-point gaussians (acc_e already is squared distance)
    float E0 = expf(-0.5f * a0);
    float E1 = expf(-0.5f * a1);
    float E2 = expf(-0.5f * a2);
    float E3 = expf(-0.5f * a3);

    if (isQ) {
        float amp = amplitude[0];
        float w0 = 1.f / (1.f + expf(-strengths[0])); if (!(w0 > 0.1f)) w0 = 0.f;
        float w1 = 1.f / (1.f + expf(-strengths[1])); if (!(w1 > 0.1f)) w1 = 0.f;
        float w2 = 1.f / (1.f + expf(-strengths[2])); if (!(w2 > 0.1f)) w2 = 0.f;
        float w3 = 1.f / (1.f + expf(-strengths[3])); if (!(w3 > 0.1f)) w3 = 0.f;
        float* F = Fq + (size_t)idx * NFEAT;
        *(v4f*)(F + 0) = (v4f){amp * S, amp * w0 * E0, amp * w1 * E1, amp * w2 * E2};
        *(v4f*)(F + 4) = (v4f){amp * w3 * E3, 0.f, 0.f, 0.f};
    } else {
        float* F = Fk + (size_t)idx * NFEAT;
        *(v4f*)(F + 0) = (v4f){S, E0, E1, E2};
        *(v4f*)(F + 4) = (v4f){E3, 0.f, 0.f, 0.f};
    }
}

// ---------------------------------------------------------------------------
// Phase 2: out[b] = Fq[b] (Nq x 8) * Fk[b]^T (8 x Nk), rank-8 via two
// V_WMMA_F32_16X16X4_F32 per 16x16 tile. One wave: 1 q-tile x KT_GROUP k-tiles.
// A-matrix 16x4 f32: lanes 0-15 -> K=0,1 ; lanes 16-31 -> K=2,3 (float2/lane).
// B-matrix 4x16 f32: same striping with N = lane%16.
// C/D: VGPR r holds (row r + 8*(lane/16), col lane%16).
// ---------------------------------------------------------------------------
__global__ __launch_bounds__(256) void hts_rank8_gemm(
    const float* __restrict__ Fq, const float* __restrict__ Fk,
    float* __restrict__ out)
{
    const int nkg   = NK / (16 * KT_GROUP);            // k-tile groups per row
    const int wave  = blockIdx.x * (blockDim.x >> 5) + (threadIdx.x >> 5);
    const int lane  = threadIdx.x & 31;
    const int half  = lane >> 4;                       // 0: lanes 0-15, 1: 16-31
    const int l16   = lane & 15;

    const int kg = wave % nkg;
    const int qt = (wave / nkg) % (NQ / 16);
    const int b  = wave / (nkg * (NQ / 16));

    // A fragments: feature row of query (qt*16 + l16)
    const float* __restrict__ qrow = Fq + ((size_t)b * NQ + qt * 16 + l16) * NFEAT;
    v2f afrag0 = *(const v2f*)(qrow + 2 * half);       // K = 0..3
    v2f afrag1 = *(const v2f*)(qrow + 4 + 2 * half);   // K = 4..7

    const float* __restrict__ fkb = Fk + (size_t)b * NK * NFEAT;
    float* __restrict__ obase =
        out + ((size_t)b * NQ + (size_t)qt * 16) * NK;

    int kt0 = kg * KT_GROUP;
#pragma unroll 4
    for (int t = 0; t < KT_GROUP; ++t) {
        int kt = kt0 + t;
        const float* __restrict__ krow = fkb + ((size_t)kt * 16 + l16) * NFEAT;
        v2f bfrag0 = *(const v2f*)(krow + 2 * half);
        v2f bfrag1 = *(const v2f*)(krow + 4 + 2 * half);

        v8f c = {};
        // 8 args: (neg_a, A, neg_b, B, c_mod, C, reuse_a, reuse_b)
        c = __builtin_amdgcn_wmma_f32_16x16x4_f32(false, afrag0, false, bfrag0,
                                                  (short)0, c, false, false);
        c = __builtin_amdgcn_wmma_f32_16x16x4_f32(false, afrag1, false, bfrag1,
                                                  (short)0, c, false, false);

        float* __restrict__ o = obase + (size_t)kt * 16 + l16;
#pragma unroll
        for (int r = 0; r < 8; ++r) {
            int row = r + 8 * half;
            __builtin_nontemporal_store(c[r], o + (size_t)row * NK);
        }
    }
}

extern "C" void kernel_launch(void* const* d_in, const int* in_sizes, int n_in,
                              void* d_out, int out_size, void* d_ws, size_t ws_size,
                              hipStream_t stream) {
    const float* queries   = (const float*)d_in[0];  // [B,NQ,DIM]
    const float* keys      = (const float*)d_in[1];  // [B,NK,DIM]
    const float* center    = (const float*)d_in[2];  // [DIM]
    const float* outer_r   = (const float*)d_in[3];  // [1]
    const float* hole_r    = (const float*)d_in[4];  // [1]
    const float* entry     = (const float*)d_in[5];  // [E,DIM]
    const float* strengths = (const float*)d_in[6];  // [E]
    const float* amplitude = (const float*)d_in[7];  // [1]
    float* out = (float*)d_out;

    float* Fq = (float*)d_ws;                                  // B*NQ*8 f32 = 512 KB
    float* Fk = Fq + (size_t)BB * NQ * NFEAT;                  // B*NK*8 f32 = 512 KB

    {
        int total  = BB * NQ + BB * NK;
        int blocks = (total + 255) / 256;
        hts_features<<<blocks, 256, 0, stream>>>(queries, keys, center, outer_r,
                                                 hole_r, entry, strengths,
                                                 amplitude, Fq, Fk);
    }
    {
        int waves  = BB * (NQ / 16) * (NK / (16 * KT_GROUP));  // 16384
        int blocks = waves / 8;                                // 8 waves / block
        hts_rank8_gemm<<<blocks, 256, 0, stream>>>(Fq, Fk, out);
    }
}